// PostProcess_1958505087755
// MI455X (gfx1250) — compile-verified
//
#include <hip/hip_runtime.h>
#include <hip/hip_bf16.h>
#include <stdint.h>

// ---------------- problem constants ----------------
#define NIMG   16
#define NANCH  25200
#define NCLS   80
#define KPRE   2048
#define CAP    8192      // compaction capacity per image (>= KPRE + tie-bucket slack)
#define NDET   300
#define APB    120       // anchors per block (divides 19200 / 4800 / 1200)
#define TPB    128
#define CHUNKS 210       // 160 (lvl0) + 40 (lvl1) + 10 (lvl2) chunks per image

__device__ __forceinline__ float sigm(float x) {
  return 1.0f / (1.0f + __expf(-x));
}

// CDNA5 async data mover: stage ndw contiguous dwords global -> LDS using
// 16-byte async transfers (512 B per wave per instruction). Per ISA
// CDNA5 ISA: each lane copies 16 B from its global address to its LDS address,
// tracked by ASYNCcnt; fence with s_wait_asynccnt then workgroup barrier.
// Low 32 bits of a flat shared-memory address == LDS byte offset.
// Caller guarantees 16-B alignment of g and smem, and ndw % 4 == 0.
__device__ __forceinline__ void async_stage_b128(const float* __restrict__ g,
                                                 float* smem, int ndw) {
  unsigned lds0 = (unsigned)(size_t)smem;
  int nquad = ndw >> 2;
  for (int q = threadIdx.x; q < nquad; q += blockDim.x) {
    unsigned     lds = lds0 + (unsigned)q * 16u;
    const float* ga  = g + q * 4;
    asm volatile("global_load_async_to_lds_b128 %0, %1, off"
                 :: "v"(lds), "v"(ga) : "memory");
  }
  asm volatile("s_wait_asynccnt 0x0" ::: "memory");
  __syncthreads();
}

// ---------------- pass kernel: decode + histogram (PASS=0) / compact (PASS=1) --
template <int PASS>
__global__ __launch_bounds__(TPB) void pass_kernel(
    const float* __restrict__ h0, const float* __restrict__ h1,
    const float* __restrict__ h2, const float* __restrict__ grids,
    const float* __restrict__ strides, const float* __restrict__ anch,
    float* __restrict__ boxes, unsigned* __restrict__ hist,
    const unsigned* __restrict__ Tb, unsigned* __restrict__ cnt,
    unsigned long long* __restrict__ keys) {
  __shared__ alignas(16) float sm[APB * 85];   // 40800 B, 16-B aligned tile

  int img = blockIdx.x / CHUNKS;
  int cid = blockIdx.x - img * CHUNKS;
  const float* hp;
  int lvl_off;
  if (cid < 160) {                   // level 0: 3*80*80 anchors, 1,632,000 floats/img
    hp = h0 + (size_t)img * 1632000 + (size_t)cid * (APB * 85);
    lvl_off = cid * APB;
  } else if (cid < 200) {            // level 1: 3*40*40 anchors, 408,000 floats/img
    int c = cid - 160;
    hp = h1 + (size_t)img * 408000 + (size_t)c * (APB * 85);
    lvl_off = 19200 + c * APB;
  } else {                           // level 2: 3*20*20 anchors, 102,000 floats/img
    int c = cid - 200;
    hp = h2 + (size_t)img * 102000 + (size_t)c * (APB * 85);
    lvl_off = 24000 + c * APB;
  }

  async_stage_b128(hp, sm, APB * 85);   // APB*85 = 10200 dwords = 2550 x b128

  int t = threadIdx.x;
  if (t < APB) {
    int n = lvl_off + t;
    const float* L = sm + t * 85;
    float obj = sigm(L[4]);

    if (PASS == 0) {
      float gx = grids[n * 2], gy = grids[n * 2 + 1];
      float st = strides[n * 2];
      float aw = anch[n * 2], ah = anch[n * 2 + 1];
      float sx = sigm(L[0]), sy = sigm(L[1]);
      float sw = sigm(L[2]), sh = sigm(L[3]);
      float cx = (sx * 2.f - 0.5f + gx) * st;
      float cy = (sy * 2.f - 0.5f + gy) * st;
      float ww = (sw * 2.f) * (sw * 2.f) * aw;
      float hh = (sh * 2.f) * (sh * 2.f) * ah;
      float4 bx = make_float4(cx - ww * 0.5f, cy - hh * 0.5f,
                              cx + ww * 0.5f, cy + hh * 0.5f);
      ((float4*)boxes)[img * NANCH + n] = bx;
    }

    unsigned T = (PASS == 1) ? Tb[img] : 0u;
    for (int c = 0; c < NCLS; ++c) {
      float s = obj * sigm(L[5 + c]);
      if (s > 0.25f) {
        // score in (0.25, 1) -> float bits in [0x3E800000, 0x3F800000): 2048 buckets
        unsigned bkt = (__float_as_uint(s) - 0x3E800000u) >> 13;
        if (PASS == 0) {
          atomicAdd(&hist[img * 2048 + bkt], 1u);
        } else if (bkt >= T) {
          unsigned pos = atomicAdd(&cnt[img], 1u);
          if (pos < CAP) {
            unsigned idx = (unsigned)(n * NCLS + c);
            keys[(size_t)img * CAP + pos] =
                ((unsigned long long)__float_as_uint(s) << 32) |
                (unsigned long long)(0xFFFFFFFFu - idx);
          }
        }
      }
    }
  }
}

// ---------------- per-image threshold-bucket scan -----------------------------
__global__ void scan_kernel(const unsigned* __restrict__ hist,
                            unsigned* __restrict__ Tb) {
  int b = threadIdx.x;
  if (b >= NIMG) return;
  unsigned cum = 0, T = 0;
  bool found = false;
  for (int k = 2047; k >= 0; --k) {
    cum += hist[b * 2048 + k];
    if (!found && cum >= KPRE) { T = (unsigned)k; found = true; }
  }
  Tb[b] = T;  // found==false (total < 2048) -> T=0 -> compact everything
}

// ---------------- per-image bitonic sort of 8192 keys (desc) ------------------
__global__ __launch_bounds__(512) void sort_kernel(
    const unsigned long long* __restrict__ keys, const unsigned* __restrict__ cnt,
    unsigned long long* __restrict__ top, unsigned* __restrict__ validn) {
  __shared__ alignas(16) unsigned long long sk[CAP];  // 64 KB LDS
  int b = blockIdx.x;
  unsigned m = cnt[b];
  if (m > CAP) m = CAP;
  for (int i = threadIdx.x; i < CAP; i += 512)
    sk[i] = (i < (int)m) ? keys[(size_t)b * CAP + i] : 0ull;
  __syncthreads();
  for (int k = 2; k <= CAP; k <<= 1) {
    for (int j = k >> 1; j > 0; j >>= 1) {
      for (int i = threadIdx.x; i < CAP; i += 512) {
        int ixj = i ^ j;
        if (ixj > i) {
          unsigned long long a = sk[i], c = sk[ixj];
          bool asc = (i & k) == 0;
          if (asc ? (a < c) : (a > c)) { sk[i] = c; sk[ixj] = a; }  // descending
        }
      }
      __syncthreads();
    }
  }
  for (int i = threadIdx.x; i < KPRE; i += 512)
    top[(size_t)b * KPRE + i] = sk[i];
  if (threadIdx.x == 0) {
    unsigned v = cnt[b];
    if (v > KPRE) v = KPRE;
    validn[b] = v;
  }
}

// ---------------- gather candidate boxes/scores/labels ------------------------
__global__ __launch_bounds__(256) void gather_kernel(
    const unsigned long long* __restrict__ top, const unsigned* __restrict__ validn,
    const float* __restrict__ boxes, float* __restrict__ cbox,
    float* __restrict__ cscore, int* __restrict__ clab) {
  int g = blockIdx.x * 256 + threadIdx.x;
  if (g >= NIMG * KPRE) return;
  int b = g / KPRE, r = g - b * KPRE;
  unsigned long long key = top[g];
  float4 bx = make_float4(0.f, 0.f, 0.f, 0.f);
  float s = 0.f;
  int lab = 0;
  if (r < (int)validn[b] && key != 0ull) {
    unsigned idx = 0xFFFFFFFFu - (unsigned)(key & 0xFFFFFFFFull);
    s = __uint_as_float((unsigned)(key >> 32));
    int n = (int)(idx / NCLS);
    lab = (int)(idx - (unsigned)n * NCLS);
    bx = ((const float4*)boxes)[b * NANCH + n];
  }
  ((float4*)cbox)[g] = bx;
  cscore[g] = s;
  clab[g] = lab;
}

// ---------------- per-image greedy NMS + top-300 emit -------------------------
__global__ __launch_bounds__(256) void nms_kernel(
    const float* __restrict__ cbox, const float* __restrict__ cscore,
    const int* __restrict__ clab, const unsigned* __restrict__ validn,
    float* __restrict__ out_boxes, float* __restrict__ out_scores,
    int* __restrict__ out_labels) {
  __shared__ float X1[KPRE], Y1[KPRE], X2[KPRE], Y2[KPRE], AR[KPRE];
  __shared__ unsigned sup[KPRE / 32];
  __shared__ int s_keep, s_flag, s_stop;
  int b = blockIdx.x;
  int nv = (int)validn[b];

  for (int r = threadIdx.x; r < KPRE; r += 256) {
    float4 bx = ((const float4*)cbox)[b * KPRE + r];
    float off = (float)clab[b * KPRE + r] * 4096.0f;  // batched-NMS class offset
    float x1 = bx.x + off, y1 = bx.y + off, x2 = bx.z + off, y2 = bx.w + off;
    X1[r] = x1; Y1[r] = y1; X2[r] = x2; Y2[r] = y2;
    AR[r] = fmaxf(x2 - x1, 0.f) * fmaxf(y2 - y1, 0.f);
  }
  for (int r = threadIdx.x; r < KPRE / 32; r += 256) sup[r] = 0u;
  if (threadIdx.x == 0) { s_keep = 0; s_stop = 0; }
  __syncthreads();

  for (int i = 0; i < nv; ++i) {
    if (threadIdx.x == 0) {
      bool supd = (sup[i >> 5] >> (i & 31)) & 1u;
      s_flag = supd ? 0 : 1;
      if (!supd) {
        int slot = s_keep++;
        float4 bx = ((const float4*)cbox)[b * KPRE + i];
        out_boxes[(b * NDET + slot) * 4 + 0] = bx.x;
        out_boxes[(b * NDET + slot) * 4 + 1] = bx.y;
        out_boxes[(b * NDET + slot) * 4 + 2] = bx.z;
        out_boxes[(b * NDET + slot) * 4 + 3] = bx.w;
        out_scores[b * NDET + slot] = cscore[b * KPRE + i];
        out_labels[b * NDET + slot] = clab[b * KPRE + i];
        if (s_keep >= NDET) s_stop = 1;  // keeps past #300 are discarded anyway
      }
    }
    __syncthreads();
    if (s_stop) break;
    if (s_flag) {
      float ax1 = X1[i], ay1 = Y1[i], ax2 = X2[i], ay2 = Y2[i], aar = AR[i];
      for (int j = i + 1 + threadIdx.x; j < nv; j += 256) {
        float xx1 = fmaxf(ax1, X1[j]), yy1 = fmaxf(ay1, Y1[j]);
        float xx2 = fminf(ax2, X2[j]), yy2 = fminf(ay2, Y2[j]);
        float w = fmaxf(xx2 - xx1, 0.f), h = fmaxf(yy2 - yy1, 0.f);
        float inter = w * h;
        float iou = inter / (aar + AR[j] - inter + 1e-7f);
        if (iou > 0.45f) atomicOr(&sup[j >> 5], 1u << (j & 31));
      }
    }
    __syncthreads();
  }

  int kc = s_keep;
  if (kc > NDET) kc = NDET;
  for (int slot = kc + threadIdx.x; slot < NDET; slot += 256) {
    out_boxes[(b * NDET + slot) * 4 + 0] = 0.f;
    out_boxes[(b * NDET + slot) * 4 + 1] = 0.f;
    out_boxes[(b * NDET + slot) * 4 + 2] = 0.f;
    out_boxes[(b * NDET + slot) * 4 + 3] = 0.f;
    out_scores[b * NDET + slot] = 0.f;
    out_labels[b * NDET + slot] = -1;
  }
}

// ---------------- launch ------------------------------------------------------
extern "C" void kernel_launch(void* const* d_in, const int* in_sizes, int n_in,
                              void* d_out, int out_size, void* d_ws,
                              size_t ws_size, hipStream_t stream) {
  const float* h0 = (const float*)d_in[0];
  const float* h1 = (const float*)d_in[1];
  const float* h2 = (const float*)d_in[2];
  const float* grids = (const float*)d_in[3];
  const float* strides = (const float*)d_in[4];
  const float* anch = (const float*)d_in[5];

  // workspace layout (bytes), ~8.7 MB total
  char* ws = (char*)d_ws;
  float*              boxes  = (float*)(ws + 0);                  // 16*25200*4 f32
  unsigned*           hist   = (unsigned*)(ws + 6451200);         // 16*2048 u32
  unsigned*           Tb     = (unsigned*)(ws + 6582272);         // 16 u32
  unsigned*           cnt    = (unsigned*)(ws + 6582336);         // 16 u32
  unsigned long long* keys   = (unsigned long long*)(ws + 6582400);  // 16*8192 u64
  unsigned long long* top    = (unsigned long long*)(ws + 7630976);  // 16*2048 u64
  float*              cbox   = (float*)(ws + 7893120);            // 16*2048*4 f32
  float*              cscore = (float*)(ws + 8417408);            // 16*2048 f32
  int*                clab   = (int*)(ws + 8548480);              // 16*2048 i32
  unsigned*           validn = (unsigned*)(ws + 8679552);         // 16 u32

  float* out_boxes  = (float*)d_out;                  // [16,300,4]
  float* out_scores = out_boxes + NIMG * NDET * 4;    // [16,300]
  int*   out_labels = (int*)(out_scores + NIMG * NDET);  // [16,300] int32 bits

  hipMemsetAsync(hist, 0, NIMG * 2048 * sizeof(unsigned), stream);
  hipMemsetAsync(cnt, 0, NIMG * sizeof(unsigned), stream);

  pass_kernel<0><<<NIMG * CHUNKS, TPB, 0, stream>>>(
      h0, h1, h2, grids, strides, anch, boxes, hist, Tb, cnt, keys);
  scan_kernel<<<1, 32, 0, stream>>>(hist, Tb);
  pass_kernel<1><<<NIMG * CHUNKS, TPB, 0, stream>>>(
      h0, h1, h2, grids, strides, anch, boxes, hist, Tb, cnt, keys);
  sort_kernel<<<NIMG, 512, 0, stream>>>(keys, cnt, top, validn);
  gather_kernel<<<(NIMG * KPRE + 255) / 256, 256, 0, stream>>>(
      top, validn, boxes, cbox, cscore, clab);
  nms_kernel<<<NIMG, 256, 0, stream>>>(cbox, cscore, clab, validn, out_boxes,
                                       out_scores, out_labels);
}